// RegionalFocalLoss_57982058496539
// MI455X (gfx1250) — compile-verified
//
#include <hip/hip_runtime.h>
#include <cstdint>

// ---------------------------------------------------------------------------
// Regional focal loss, MI455X (gfx1250).
// Stage 1: TDM double-buffered stream (global -> LDS) + ds_add_f32 segment
//          accumulation into per-block LDS tables, flushed as partial slices.
// Stage 2: column reduction over partial slices (in place at ws[0..ACC)).
// Stage 3: single-block 117-region finalize -> scalar loss.
// ---------------------------------------------------------------------------

#define REGIONS   117
#define NBINS     64
#define SUM_OFF   0
#define INV_OFF   REGIONS
#define HIST_OFF  (2 * REGIONS)
#define ACC       (2 * REGIONS + REGIONS * NBINS)   // 7722 floats
#define TILE      1024                              // elements per TDM tile
#define THREADS   256
#define MAXPART   512
#define HWDIV     262144.0f                         // H*W = 512*512
#define ALPHA_C   0.07f
#define BETA_C    0.03f
#define LAM_C     1.0f
#define EPS_C     1e-6f

typedef uint32_t v4u __attribute__((ext_vector_type(4)));
typedef int      v8i __attribute__((ext_vector_type(8)));
typedef int      v4i __attribute__((ext_vector_type(4)));

// Issue one TDM load of TILE consecutive 4-byte elements into LDS.
// Descriptor layout per CDNA5 ISA sec 8.3/8.4 (D# groups 0 and 1).
__device__ __forceinline__ void tdm_tile_load(uint32_t lds_byte_off,
                                              const void* base,
                                              unsigned long long elem_off) {
  unsigned long long ga = (unsigned long long)(uintptr_t)base + elem_off * 4ull;
  v4u g0;
  g0[0] = 1u;                                         // count=1, user D#
  g0[1] = lds_byte_off;                               // lds_addr
  g0[2] = (uint32_t)ga;                               // global_addr[31:0]
  g0[3] = ((uint32_t)(ga >> 32) & 0x01FFFFFFu)        // global_addr[56:32]
          | (2u << 30);                               // type=2 ("image")
  v8i g1;
  g1[0] = 2 << 16;                                    // data_size=4B
  g1[1] = (TILE & 0xFFFF) << 16;                      // tensor_dim0 lo16
  g1[2] = ((TILE >> 16) & 0xFFFF) | (1 << 16);        // dim0 hi16 | tensor_dim1=1
  g1[3] = (TILE & 0xFFFF) << 16;                      // tile_dim0
  g1[4] = 1;                                          // tile_dim1=1, tile_dim2=0
  g1[5] = TILE;                                       // tensor_dim0_stride lo32
  g1[6] = 0;
  g1[7] = 0;
  v4i z4 = {0, 0, 0, 0};
  v8i z8 = {0, 0, 0, 0, 0, 0, 0, 0};
  __builtin_amdgcn_tensor_load_to_lds(g0, g1, z4, z4, z8, 0);
}

__device__ __forceinline__ void accum_elem(float x, float f, int reg,
                                           float* __restrict__ acc) {
  float err = fabsf(x - f);
  atomicAdd(&acc[SUM_OFF + reg], err);                // ds_add_f32
  int bin = (int)floorf(x * (float)NBINS);
  bin = bin < 0 ? 0 : (bin > NBINS - 1 ? NBINS - 1 : bin);
  if (x >= 0.0f && x <= 1.0f)
    atomicAdd(&acc[HIST_OFF + reg * NBINS + bin], 1.0f);
  else
    atomicAdd(&acc[INV_OFF + reg], 1.0f);
}

__global__ __launch_bounds__(THREADS)
void rfl_partials_kernel(const float* __restrict__ real,
                         const float* __restrict__ fake,
                         const int*   __restrict__ rmap,
                         float* __restrict__ ws,
                         long long N, long long nTiles) {
  __shared__ float s_acc[ACC];
  __shared__ float s_bufR[2][TILE];
  __shared__ float s_bufF[2][TILE];
  __shared__ int   s_bufM[2][TILE];

  const int tid = threadIdx.x;
  const bool isW0 = (tid < 32);     // wave 0 owns TDM issue + tensorcnt

  for (int j = tid; j < ACC; j += THREADS) s_acc[j] = 0.0f;

  // Prologue: wave 0 issues tile0 loads into buffer 0.
  long long t0 = blockIdx.x;
  if (isW0 && t0 < nTiles) {
    unsigned long long eoff = (unsigned long long)t0 * TILE;
    tdm_tile_load((uint32_t)(uintptr_t)&s_bufR[0][0], real, eoff);
    tdm_tile_load((uint32_t)(uintptr_t)&s_bufF[0][0], fake, eoff);
    tdm_tile_load((uint32_t)(uintptr_t)&s_bufM[0][0], rmap, eoff);
  }

  int parity = 0;
  for (long long t = t0; t < nTiles; t += gridDim.x) {
    if (isW0) __builtin_amdgcn_s_wait_tensorcnt(0);   // tile t landed in LDS
    __syncthreads();                                  // ...visible to all waves
    long long tn = t + gridDim.x;
    if (isW0 && tn < nTiles) {                        // prefetch next tile
      int nb = parity ^ 1;
      unsigned long long eoff = (unsigned long long)tn * TILE;
      tdm_tile_load((uint32_t)(uintptr_t)&s_bufR[nb][0], real, eoff);
      tdm_tile_load((uint32_t)(uintptr_t)&s_bufF[nb][0], fake, eoff);
      tdm_tile_load((uint32_t)(uintptr_t)&s_bufM[nb][0], rmap, eoff);
    }
    // Process current buffer: 256 threads x one float4/int4 group = TILE elems.
    {
      const float4* R = (const float4*)&s_bufR[parity][0];
      const float4* F = (const float4*)&s_bufF[parity][0];
      const int4*   M = (const int4*)&s_bufM[parity][0];
      for (int j = tid; j < TILE / 4; j += THREADS) {
        float4 r = R[j];
        float4 f = F[j];
        int4   m = M[j];
        accum_elem(r.x, f.x, m.x, s_acc);
        accum_elem(r.y, f.y, m.y, s_acc);
        accum_elem(r.z, f.z, m.z, s_acc);
        accum_elem(r.w, f.w, m.w, s_acc);
      }
    }
    parity ^= 1;
  }

  // Tail elements (N not a multiple of TILE): block 0, plain global loads.
  if (blockIdx.x == 0) {
    for (long long i = nTiles * TILE + tid; i < N; i += THREADS)
      accum_elem(real[i], fake[i], rmap[i], s_acc);
  }

  __syncthreads();
  // Flush this block's partial slice (no global atomics).
  float* part = ws + (size_t)blockIdx.x * ACC;
  for (int j = tid; j < ACC; j += THREADS) part[j] = s_acc[j];
}

__global__ __launch_bounds__(THREADS)
void rfl_reduce_kernel(float* __restrict__ ws, int nPart) {
  int j = blockIdx.x * blockDim.x + threadIdx.x;
  if (j >= ACC) return;
  float a = 0.0f;
  for (int b = 0; b < nPart; ++b) a += ws[(size_t)b * ACC + j];
  ws[j] = a;   // in place: column j is read/written only by this thread
}

__global__ __launch_bounds__(128)
void rfl_finalize_kernel(const float* __restrict__ red,
                         float* __restrict__ out) {
  __shared__ float s_rec[128];
  __shared__ float s_ent[128];
  const int t = threadIdx.x;

  float rec = -1e30f, ent = -1e30f, rsum = 0.0f;
  if (t < REGIONS) {
    rsum = red[SUM_OFF + t];
    float inval = red[INV_OFF + t];
    const float* h = red + HIST_OFF + t * NBINS;
    float tot = 0.0f;
    for (int b = 0; b < NBINS; ++b) tot += h[b];
    float cnt = tot + inval;                      // total element count
    rec = (cnt > 0.0f) ? (rsum / cnt) : 0.0f;     // per-region mean L1
    float inv = 1.0f / (tot + EPS_C);
    float e = 0.0f;
    for (int b = 0; b < NBINS; ++b) {
      float p = h[b] * inv;
      e -= p * logf(p + EPS_C);
    }
    ent = e;
  }
  s_rec[t] = rec;
  s_ent[t] = ent;
  __syncthreads();
  for (int s = 64; s > 0; s >>= 1) {
    if (t < s) {
      s_rec[t] = fmaxf(s_rec[t], s_rec[t + s]);
      s_ent[t] = fmaxf(s_ent[t], s_ent[t + s]);
    }
    __syncthreads();
  }
  float maxrec = s_rec[0];
  float maxent = s_ent[0];
  __syncthreads();

  float contrib = 0.0f;
  if (t < REGIONS) {
    // IEEE: 0/0 -> NaN; any NaN term poisons the whole weight -> 0 (nan_to_num)
    float w = ALPHA_C * rec / maxrec + BETA_C * ent / maxent;
    if (isnan(w)) w = 0.0f;
    contrib = (1.0f + LAM_C * w) * (rsum / HWDIV);
  }
  s_rec[t] = contrib;
  __syncthreads();
  for (int s = 64; s > 0; s >>= 1) {
    if (t < s) s_rec[t] += s_rec[t + s];
    __syncthreads();
  }
  if (t == 0) out[0] = s_rec[0];
}

extern "C" void kernel_launch(void* const* d_in, const int* in_sizes, int n_in,
                              void* d_out, int out_size, void* d_ws, size_t ws_size,
                              hipStream_t stream) {
  const float* real = (const float*)d_in[0];
  const float* fake = (const float*)d_in[1];
  const int*   rmap = (const int*)d_in[2];
  // d_in[3] = regions scalar (117) -- fixed at compile time (LDS table sizes).

  long long N = (long long)in_sizes[0];
  long long nTiles = N / TILE;

  int maxPart = (int)(ws_size / (ACC * sizeof(float)));
  int nPart = maxPart < 1 ? 1 : (maxPart > MAXPART ? MAXPART : maxPart);

  float* ws = (float*)d_ws;
  rfl_partials_kernel<<<nPart, THREADS, 0, stream>>>(real, fake, rmap, ws, N, nTiles);
  rfl_reduce_kernel<<<(ACC + THREADS - 1) / THREADS, THREADS, 0, stream>>>(ws, nPart);
  rfl_finalize_kernel<<<1, 128, 0, stream>>>(ws, (float*)d_out);
}